// NewlyDefinedLoss3_5351529251096
// MI455X (gfx1250) — compile-verified
//
#include <hip/hip_runtime.h>

// -----------------------------------------------------------------------------
// DeepHit-style NLL loss, N=65536, Q=4, K=128, fp32.
//   loss_i = sum_{k<=d_i} [ logsumexp(phi[i,0..3,k], r_k) - sel_k ],
//   r_k = 1 - sum_q phi[i,q,k]; sel_k = r_k except k==d_i && e_i>0 where
//   sel = phi[i, e_i-1, d_i].  Output = mean_i loss_i (scalar f32).
//
// HBM-bound (prefix read ~64 MB). CDNA5 async global->LDS DMA, depth-2
// prefetch pipeline (3 LDS buffers/wave), per-lane EXEC masking for the
// k<=d prefix skip, wave32 shfl reductions, deterministic 2-stage sum.
// -----------------------------------------------------------------------------

#define WPB   8                 // waves per block (256 threads)
#define SPW   8                 // samples per wave
#define NBUF  3                 // pipeline depth 2 -> 3 buffers
#define BLOCK (WPB * 32)

// Async copy of 16 bytes/lane from global memory into LDS (CDNA5 path).
// VDST = LDS byte offset (workgroup-relative), VADDR = 64-bit global address.
// phi is streamed exactly once -> non-temporal hint.
__device__ __forceinline__ void async_row_load16(uint32_t lds_off, const float4* g) {
    asm volatile("global_load_async_to_lds_b128 %0, %1, off th:TH_LOAD_NT"
                 :: "v"(lds_off), "v"(g)
                 : "memory");
}

// Wait until ASYNCcnt <= n  (n in {0,4,8}; async loads retire in order).
__device__ __forceinline__ void wait_async_le(int n) {
    if (n >= 8)      asm volatile("s_wait_asynccnt 0x8" ::: "memory");
    else if (n >= 4) asm volatile("s_wait_asynccnt 0x4" ::: "memory");
    else             asm volatile("s_wait_asynccnt 0x0" ::: "memory");
}

__device__ __forceinline__ float comp4(const float4& v, int c) {
    return (c == 0) ? v.x : (c == 1) ? v.y : (c == 2) ? v.z : v.w;
}

__global__ void __launch_bounds__(BLOCK)
pc_hazard_loss_kernel(const float* __restrict__ phi,
                      const int*   __restrict__ idx_dur,
                      const int*   __restrict__ events,
                      float*       __restrict__ partials,
                      int N)
{
    // Per-wave triple-buffered stage: [wave][buf][q-row][lane] float4 = 48 KB.
    __shared__ float4 stage[WPB][NBUF][4][32];
    __shared__ float  wsum[WPB];

    const int lane = threadIdx.x & 31;
    const int w    = threadIdx.x >> 5;
    const long wave_id     = (long)blockIdx.x * WPB + w;
    const long base_sample = wave_id * SPW;

    const float4* phi4  = (const float4*)phi;   // row (i,q) = 32 float4s
    const int     kbase = lane * 4;             // this lane's first k column

    // ---- fetch all 8 (duration, event) pairs for this wave up front ----
    int my_d = 0, my_e = 0;
    {
        const long il = base_sample + lane;
        if (lane < SPW && il < N) {
            my_d = idx_dur[il];
            my_e = events[il];
        }
    }

    // issue async loads for sample t into buffer (t % NBUF)
    auto prefetch = [&](int t) {
        const long i = base_sample + t;
        if (t < SPW && i < N) {
            const int d = __shfl(my_d, t, 32);
            if (kbase <= d) {                   // prefix skip: lanes past d idle
                const float4* g = phi4 + (size_t)i * 128 + lane;
                const int buf = t % NBUF;
                #pragma unroll
                for (int q = 0; q < 4; ++q) {
                    uint32_t off = (uint32_t)(uintptr_t)(&stage[w][buf][q][lane]);
                    async_row_load16(off, g + q * 32);
                }
            }
        }
    };
    // #async instructions issued for sample t (lane 0 always active when valid)
    auto issued = [&](int t) -> int {
        return (t < SPW && base_sample + t < N) ? 4 : 0;
    };

    float acc = 0.0f;

    prefetch(0);
    prefetch(1);

    for (int j = 0; j < SPW; ++j) {
        prefetch(j + 2);                        // keep 2 samples in flight
        wait_async_le(issued(j + 1) + issued(j + 2));   // retire sample j

        const long i = base_sample + j;
        if (i < N) {
            const int d = __shfl(my_d, j, 32);
            const int e = __shfl(my_e, j, 32);
            const int buf = j % NBUF;
            if (kbase <= d) {
                const float4 a0 = stage[w][buf][0][lane];
                const float4 a1 = stage[w][buf][1][lane];
                const float4 a2 = stage[w][buf][2][lane];
                const float4 a3 = stage[w][buf][3][lane];
                #pragma unroll
                for (int c = 0; c < 4; ++c) {
                    const int k = kbase + c;
                    if (k <= d) {
                        const float x0 = comp4(a0, c);
                        const float x1 = comp4(a1, c);
                        const float x2 = comp4(a2, c);
                        const float x3 = comp4(a3, c);
                        const float r  = 1.0f - (x0 + x1 + x2 + x3);
                        float m = fmaxf(fmaxf(fmaxf(x0, x1), fmaxf(x2, x3)), r);
                        float t = __expf(x0 - m) + __expf(x1 - m) +
                                  __expf(x2 - m) + __expf(x3 - m) + __expf(r - m);
                        const float L = m + __logf(t);
                        float sel = r;
                        if (k == d && e > 0) {
                            sel = (e == 1) ? x0 : (e == 2) ? x1 : (e == 3) ? x2 : x3;
                        }
                        acc += L - sel;
                    }
                }
            }
        }
    }

    // ---- deterministic wave32 + block reduction ----
    #pragma unroll
    for (int off = 16; off > 0; off >>= 1)
        acc += __shfl_xor(acc, off, 32);
    if (lane == 0) wsum[w] = acc;
    __syncthreads();
    if (threadIdx.x == 0) {
        float s = 0.0f;
        #pragma unroll
        for (int q = 0; q < WPB; ++q) s += wsum[q];
        partials[blockIdx.x] = s;
    }
}

__global__ void __launch_bounds__(256)
pc_hazard_reduce_kernel(const float* __restrict__ partials, int nparts,
                        float* __restrict__ out, float inv_n)
{
    __shared__ double ws[8];
    double s = 0.0;
    for (int t = threadIdx.x; t < nparts; t += 256) s += (double)partials[t];
    #pragma unroll
    for (int off = 16; off > 0; off >>= 1)
        s += __shfl_xor(s, off, 32);
    const int lane = threadIdx.x & 31, w = threadIdx.x >> 5;
    if (lane == 0) ws[w] = s;
    __syncthreads();
    if (threadIdx.x == 0) {
        double tot = 0.0;
        #pragma unroll
        for (int q = 0; q < 8; ++q) tot += ws[q];
        out[0] = (float)(tot * (double)inv_n);
    }
}

extern "C" void kernel_launch(void* const* d_in, const int* in_sizes, int n_in,
                              void* d_out, int out_size, void* d_ws, size_t ws_size,
                              hipStream_t stream)
{
    const float* phi = (const float*)d_in[0];
    const int*   idx = (const int*)d_in[1];
    const int*   ev  = (const int*)d_in[2];
    const int    N   = in_sizes[1];            // 65536

    float* partials = (float*)d_ws;
    const int grid = (N + WPB * SPW - 1) / (WPB * SPW);   // 1024 blocks

    pc_hazard_loss_kernel<<<grid, BLOCK, 0, stream>>>(phi, idx, ev, partials, N);
    pc_hazard_reduce_kernel<<<1, 256, 0, stream>>>(partials, grid, (float*)d_out,
                                                   1.0f / (float)N);
}